// IterativeRecoModel_13872744366792
// MI455X (gfx1250) — compile-verified
//
#include <hip/hip_runtime.h>

typedef __attribute__((ext_vector_type(2))) float v2f;
typedef __attribute__((ext_vector_type(8))) float v8f;

#define N_ANGLES 48
#define DET_U    64
#define DET_V    64
#define NVOX     (64 * 64 * 64)
#define SINO_N   (N_ANGLES * DET_V * DET_U)

// L = 0.5*sqrt(64^2 * 3) = 32*sqrt(3)
#define HALF_DIAG 55.42562584220407f
#define RAY_STEP  (2.0f * HALF_DIAG / 127.0f)

// ---------------------------------------------------------------------------
// Kernel 1: updated_reco = x + reco  (vectorized float4, 262144 floats)
// ---------------------------------------------------------------------------
__global__ __launch_bounds__(256) void add_vol_kernel(const float* __restrict__ x,
                                                      const float* __restrict__ reco,
                                                      float* __restrict__ out) {
    int i = blockIdx.x * blockDim.x + threadIdx.x;   // 65536 float4s exactly
    const float4* x4 = (const float4*)x;
    const float4* r4 = (const float4*)reco;
    float4 a = x4[i], b = r4[i];
    float4 o;
    o.x = a.x + b.x; o.y = a.y + b.y; o.z = a.z + b.z; o.w = a.w + b.w;
    ((float4*)out)[i] = o;
}

// ---------------------------------------------------------------------------
// Trilinear sample with zero outside [0,63]^3. Branchless: clamps + select.
// ---------------------------------------------------------------------------
__device__ __forceinline__ float tri_sample(const float* __restrict__ vol,
                                            float px, float py, float pz) {
    bool valid = (px >= 0.0f) & (px <= 63.0f) &
                 (py >= 0.0f) & (py <= 63.0f) &
                 (pz >= 0.0f) & (pz <= 63.0f);
    float fx = floorf(px), fy = floorf(py), fz = floorf(pz);
    float wx = px - fx, wy = py - fy, wz = pz - fz;
    int ix0 = min(max((int)fx, 0), 63);
    int iy0 = min(max((int)fy, 0), 63);
    int iz0 = min(max((int)fz, 0), 63);
    int ix1 = min(ix0 + 1, 63);
    int iy1 = min(iy0 + 1, 63);
    int iz1 = min(iz0 + 1, 63);

    int bx0 = ix0 << 12, bx1 = ix1 << 12;
    int by0 = iy0 << 6,  by1 = iy1 << 6;

    float v000 = vol[bx0 | by0 | iz0];
    float v001 = vol[bx0 | by0 | iz1];
    float v010 = vol[bx0 | by1 | iz0];
    float v011 = vol[bx0 | by1 | iz1];
    float v100 = vol[bx1 | by0 | iz0];
    float v101 = vol[bx1 | by0 | iz1];
    float v110 = vol[bx1 | by1 | iz0];
    float v111 = vol[bx1 | by1 | iz1];

    float c00 = v000 + wz * (v001 - v000);
    float c01 = v010 + wz * (v011 - v010);
    float c10 = v100 + wz * (v101 - v100);
    float c11 = v110 + wz * (v111 - v110);
    float c0  = c00 + wy * (c01 - c00);
    float c1  = c10 + wy * (c11 - c10);
    float val = c0  + wx * (c1  - c0);
    return valid ? val : 0.0f;
}

// ---------------------------------------------------------------------------
// Kernel 2: cone-beam projection. One wave32 handles 16 rays (a 16-wide u
// tile). Per WMMA iteration each lane supplies 2 samples forming the 16x4
// f32 A-matrix (M=lane&15, K-halves split across lane halves per the ISA
// layout); B = ones so V_WMMA_F32_16X16X4_F32 row-sums 4 steps into the f32
// accumulator. 32 iterations cover all 128 steps.
// ---------------------------------------------------------------------------
__global__ __launch_bounds__(256) void cone_proj_kernel(const float* __restrict__ vol,
                                                        const float* __restrict__ angles,
                                                        float* __restrict__ sino) {
    const int lane  = threadIdx.x;            // 0..31 (wave32)
    const int w     = blockIdx.x * 8 + threadIdx.y;   // global wave id
    const int m     = lane & 15;              // ray index within tile (M row)
    const int khalf = (lane >> 4) << 1;       // 0 for lanes 0-15, 2 for 16-31

    const int u_base = (w & 3) << 4;          // 4 u-tiles of 16 per v-row
    const int v      = (w >> 2) & 63;
    const int aIdx   = w >> 8;                // 256 waves per angle

    float sa, ca;
    sincosf(angles[aIdx], &sa, &ca);

    const float cu = ((float)(u_base + m) - 31.5f) * 2.0f;
    const float cv = ((float)v - 31.5f) * 2.0f;

    // ray direction = normalize(det_pix - src); det_c - src = -SDD*(c,s,0)
    float rx = -1000.0f * ca - cu * sa;
    float ry = -1000.0f * sa + cu * ca;
    float rz = cv;
    float inv = 1.0f / sqrtf(rx * rx + ry * ry + rz * rz);
    float dx = rx * inv, dy = ry * inv, dz = rz * inv;

    // start point for this lane's K-half: src + center + t0*d
    const float t0 = (500.0f - HALF_DIAG) + (float)khalf * RAY_STEP;
    float px = 500.0f * ca + 31.5f + t0 * dx;
    float py = 500.0f * sa + 31.5f + t0 * dy;
    float pz = 31.5f + t0 * dz;

    const float sdx = RAY_STEP * dx, sdy = RAY_STEP * dy, sdz = RAY_STEP * dz;
    const float jx = 4.0f * sdx, jy = 4.0f * sdy, jz = 4.0f * sdz;

    v2f bmat; bmat.x = 1.0f; bmat.y = 1.0f;   // B = ones(4x16) -> row-sum
    v8f acc = {};                             // C/D 16x16 f32 accumulator

    #pragma unroll 4
    for (int kk = 0; kk < 32; ++kk) {
        v2f amat;
        amat.x = tri_sample(vol, px, py, pz);
        amat.y = tri_sample(vol, px + sdx, py + sdy, pz + sdz);
        // D = A(16x4) * ones(4x16) + C  : accumulates 4 steps per ray
        acc = __builtin_amdgcn_wmma_f32_16x16x4_f32(
            /*neg_a=*/false, amat, /*neg_b=*/false, bmat,
            /*c_mod=*/(short)0, acc, /*reuse_a=*/false, /*reuse_b=*/false);
        px += jx; py += jy; pz += jz;
    }

    // D layout (32-bit 16x16): VGPR j -> row M=j on lanes 0-15 (N=lane),
    // row M=8+j on lanes 16-31 (N=lane-16). All columns identical (B=1),
    // so lane 0 carries rows 0..7 and lane 16 carries rows 8..15.
    const int rowbase = (aIdx * DET_V + v) * DET_U + u_base;
    if (lane == 0) {
        #pragma unroll
        for (int j = 0; j < 8; ++j) sino[rowbase + j] = acc[j] * RAY_STEP;
    } else if (lane == 16) {
        #pragma unroll
        for (int j = 0; j < 8; ++j) sino[rowbase + 8 + j] = acc[j] * RAY_STEP;
    }
}

extern "C" void kernel_launch(void* const* d_in, const int* in_sizes, int n_in,
                              void* d_out, int out_size, void* d_ws, size_t ws_size,
                              hipStream_t stream) {
    const float* x      = (const float*)d_in[0];
    const float* reco   = (const float*)d_in[1];
    const float* angles = (const float*)d_in[2];

    float* sino  = (float*)d_out;          // [48,64,64]
    float* ureco = sino + SINO_N;          // [64,64,64]

    // 262144 floats = 65536 float4 -> 256 blocks x 256 threads
    add_vol_kernel<<<256, 256, 0, stream>>>(x, reco, ureco);

    // 196608 rays / 16 rays-per-wave = 12288 waves; 8 waves per block
    cone_proj_kernel<<<1536, dim3(32, 8), 0, stream>>>(ureco, angles, sino);
}